// HeteroGraphSAGE_49950469652729
// MI455X (gfx1250) — compile-verified
//
#include <hip/hip_runtime.h>

typedef float v2f __attribute__((ext_vector_type(2)));
typedef float v8f __attribute__((ext_vector_type(8)));

// ---------------------------------------------------------------------------
// Edge-degree count: one thread per edge. From the "treats" edge list:
// cnt_drug[s]++ (drug is dst of "treated"), cnt_dis[d]++ (disease is dst of
// "treats").
// ---------------------------------------------------------------------------
__global__ void count_kernel(const int* __restrict__ src, const int* __restrict__ dst,
                             float* __restrict__ cnt_src_side, float* __restrict__ cnt_dst_side,
                             int E) {
    int e = blockIdx.x * blockDim.x + threadIdx.x;
    if (e < E) {
        atomicAdd(&cnt_src_side[src[e]], 1.0f);
        atomicAdd(&cnt_dst_side[dst[e]], 1.0f);
    }
}

// ---------------------------------------------------------------------------
// Scatter-add aggregation: (dim/4) threads per edge, each moving a float4.
// agg[dst[e]][c] += feat[src[e]][c]. dim = 4 << dim4shift.
// Source node tables live in the 192MB L2, so the random gathers mostly hit
// L2; atomics resolve in L2 as well.
// ---------------------------------------------------------------------------
__global__ void scatter_kernel(const int* __restrict__ srcIdx, const int* __restrict__ dstIdx,
                               const float* __restrict__ feat, float* __restrict__ agg,
                               int E, int dim4shift) {
    long long gid = (long long)blockIdx.x * blockDim.x + threadIdx.x;
    int e = (int)(gid >> dim4shift);
    if (e >= E) return;
    int c4  = (int)(gid & ((1 << dim4shift) - 1));
    int dim = 4 << dim4shift;
    int s = srcIdx[e];
    int d = dstIdx[e];
    const float4 v = *(const float4*)(feat + (long long)s * dim + c4 * 4);
    float* o = agg + (long long)d * dim + c4 * 4;
    atomicAdd(o + 0, v.x);
    atomicAdd(o + 1, v.y);
    atomicAdd(o + 2, v.z);
    atomicAdd(o + 3, v.w);
}

// ---------------------------------------------------------------------------
// Fused SAGE output: out = act( mean(A1sum) @ W1^T + bias + A2 @ W2^T )
// One wave32 per 16x16 output tile, fp32 WMMA (V_WMMA_F32_16X16X4_F32).
//
// Mean is folded into the A1 load path: scale = 1/max(cnt[row],1) is computed
// once per tile per lane (row fixed per lane) and applied to each loaded v2f,
// eliminating a separate full-buffer normalize pass.
//
// A-matrix VGPR layout (32-bit A 16x4): lanes 0-15 hold K=k..k+1, lanes 16-31
// hold K=k+2..k+3 -> one float2 load per lane per step. B (4x16) mirrors that
// with W[nBase+l][k+kq] (W is row-major [N,K], i.e. applied as W^T).
// C/D layout: VGPR j -> (M=j, N=lane) lanes 0-15; (M=j+8, N=lane-16) 16-31.
// ---------------------------------------------------------------------------
__global__ void sage_gemm_kernel(const float* __restrict__ A1, int lda1, int K1,
                                 const float* __restrict__ W1,
                                 const float* __restrict__ cntA1,
                                 const float* __restrict__ A2, int lda2, int K2,
                                 const float* __restrict__ W2,
                                 const float* __restrict__ bias,
                                 float* __restrict__ out, int ldo,
                                 int tilesN, int totalTiles, int doRelu) {
    int wave = blockIdx.x * (blockDim.x >> 5) + (threadIdx.x >> 5);
    if (wave >= totalTiles) return;            // wave-uniform: EXEC stays all-ones
    int lane = threadIdx.x & 31;
    int tm = wave / tilesN;
    int tn = wave - tm * tilesN;
    int rowBase = tm << 4;
    int nBase   = tn << 4;
    int lh = lane & 15;            // row (A) / out-col (B) within tile
    int kq = (lane >> 4) << 1;     // K sub-offset: 0 for lanes 0-15, 2 for 16-31

    v8f acc;
    float b = bias[nBase + lh];    // bias depends only on N -> same in all 8 VGPRs
    #pragma unroll
    for (int j = 0; j < 8; ++j) acc[j] = b;

    {
        // per-row mean scale (degree is per output row, fixed for this tile)
        float sc = 1.0f / fmaxf(cntA1[rowBase + lh], 1.0f);
        const float* aRow = A1 + (long long)(rowBase + lh) * lda1 + kq;
        const float* wRow = W1 + (long long)(nBase + lh) * K1 + kq;
        for (int k = 0; k < K1; k += 4) {
            v2f a = *(const v2f*)(aRow + k);
            v2f w = *(const v2f*)(wRow + k);
            a *= sc;
            acc = __builtin_amdgcn_wmma_f32_16x16x4_f32(false, a, false, w,
                                                        (short)0, acc, false, false);
        }
    }
    {
        const float* aRow = A2 + (long long)(rowBase + lh) * lda2 + kq;
        const float* wRow = W2 + (long long)(nBase + lh) * K2 + kq;
        for (int k = 0; k < K2; k += 4) {
            v2f a = *(const v2f*)(aRow + k);
            v2f w = *(const v2f*)(wRow + k);
            acc = __builtin_amdgcn_wmma_f32_16x16x4_f32(false, a, false, w,
                                                        (short)0, acc, false, false);
        }
    }
    if (doRelu) {
        #pragma unroll
        for (int j = 0; j < 8; ++j) acc[j] = fmaxf(acc[j], 0.0f);
    }
    int outRow0 = rowBase + (lane >> 4) * 8;
    int col     = nBase + lh;
    #pragma unroll
    for (int j = 0; j < 8; ++j)
        out[(long long)(outRow0 + j) * ldo + col] = acc[j];
}

// ---------------------------------------------------------------------------
extern "C" void kernel_launch(void* const* d_in, const int* in_sizes, int n_in,
                              void* d_out, int out_size, void* d_ws, size_t ws_size,
                              hipStream_t stream) {
    const float* x_drug   = (const float*)d_in[0];
    const float* x_dis    = (const float*)d_in[1];
    const int*   ei_tr    = (const int*)d_in[2];   // [2,E]: row0=drug(src), row1=disease(dst)
    // d_in[3] (ei_treated) is the same edges reversed; not needed.
    const float* l0_td_Wl = (const float*)d_in[4];
    const float* l0_td_bl = (const float*)d_in[5];
    const float* l0_td_Wr = (const float*)d_in[6];
    const float* l0_dt_Wl = (const float*)d_in[7];
    const float* l0_dt_bl = (const float*)d_in[8];
    const float* l0_dt_Wr = (const float*)d_in[9];
    const float* l1_td_Wl = (const float*)d_in[10];
    const float* l1_td_bl = (const float*)d_in[11];
    const float* l1_td_Wr = (const float*)d_in[12];
    const float* l1_dt_Wl = (const float*)d_in[13];
    const float* l1_dt_bl = (const float*)d_in[14];
    const float* l1_dt_Wr = (const float*)d_in[15];

    const int E   = in_sizes[2] / 2;
    const int ND  = 100000;   // drugs
    const int NS  = 20000;    // diseases
    const int DDR = 128;      // drug feature dim
    const int DDI = 64;       // disease feature dim
    const int HID = 128;
    const int OUT = 64;

    float* ws       = (float*)d_ws;
    float* agg_dis  = ws;                                  // [NS, 128]
    float* agg_drug = agg_dis  + (size_t)NS * HID;         // [ND, 128] (layer0 uses ld=64)
    float* h_dis    = agg_drug + (size_t)ND * HID;         // [NS, 128]
    float* h_drug   = h_dis    + (size_t)NS * HID;         // [ND, 128]
    float* cnt_dis  = h_drug   + (size_t)ND * HID;         // [NS]
    float* cnt_drug = cnt_dis  + NS;                       // [ND] (contiguous with cnt_dis)

    const int* tr_src = ei_tr;       // drug indices per edge
    const int* tr_dst = ei_tr + E;   // disease indices per edge

    float* o_drug = (float*)d_out;                         // [ND, OUT]
    float* o_dis  = (float*)d_out + (size_t)ND * OUT;      // [NS, OUT]

    // ---- degrees (shared by both layers) + layer-0 aggregation buffers ----
    hipMemsetAsync(cnt_dis, 0, sizeof(float) * (size_t)(NS + ND), stream);
    hipMemsetAsync(agg_dis, 0, sizeof(float) * (size_t)NS * HID, stream);
    hipMemsetAsync(agg_drug, 0, sizeof(float) * (size_t)ND * DDI, stream);

    count_kernel<<<(E + 255) / 256, 256, 0, stream>>>(tr_src, tr_dst, cnt_drug, cnt_dis, E);

    // layer 0 scatter: disease <- sum(drug feats, dim128); drug <- sum(disease feats, dim64)
    {
        long long tot = (long long)E * 32;   // 128/4 threads per edge
        scatter_kernel<<<(unsigned)((tot + 255) / 256), 256, 0, stream>>>(
            tr_src, tr_dst, x_drug, agg_dis, E, 5);
    }
    {
        long long tot = (long long)E * 16;   // 64/4 threads per edge
        scatter_kernel<<<(unsigned)((tot + 255) / 256), 256, 0, stream>>>(
            tr_dst, tr_src, x_dis, agg_drug, E, 4);
    }

    // layer 0 GEMMs (mean fused via cnt; ReLU)
    {
        int tilesN = HID / 16, total = (NS / 16) * tilesN;
        sage_gemm_kernel<<<(total + 7) / 8, 256, 0, stream>>>(
            agg_dis, HID, DDR, l0_td_Wl, cnt_dis,
            x_dis,  DDI, DDI, l0_td_Wr,
            l0_td_bl, h_dis, HID, tilesN, total, 1);
    }
    {
        int tilesN = HID / 16, total = (ND / 16) * tilesN;
        sage_gemm_kernel<<<(total + 7) / 8, 256, 0, stream>>>(
            agg_drug, DDI, DDI, l0_dt_Wl, cnt_drug,
            x_drug,  DDR, DDR, l0_dt_Wr,
            l0_dt_bl, h_drug, HID, tilesN, total, 1);
    }

    // ---- layer 1 aggregation (both dims = 128) ----
    hipMemsetAsync(agg_dis, 0, sizeof(float) * (size_t)NS * HID, stream);
    hipMemsetAsync(agg_drug, 0, sizeof(float) * (size_t)ND * HID, stream);
    {
        long long tot = (long long)E * 32;
        scatter_kernel<<<(unsigned)((tot + 255) / 256), 256, 0, stream>>>(
            tr_src, tr_dst, h_drug, agg_dis, E, 5);
        scatter_kernel<<<(unsigned)((tot + 255) / 256), 256, 0, stream>>>(
            tr_dst, tr_src, h_dis, agg_drug, E, 5);
    }

    // layer 1 GEMMs (mean fused via cnt; no activation) -> d_out
    {
        int tilesN = OUT / 16, total = (NS / 16) * tilesN;
        sage_gemm_kernel<<<(total + 7) / 8, 256, 0, stream>>>(
            agg_dis, HID, HID, l1_td_Wl, cnt_dis,
            h_dis,  HID, HID, l1_td_Wr,
            l1_td_bl, o_dis, OUT, tilesN, total, 0);
    }
    {
        int tilesN = OUT / 16, total = (ND / 16) * tilesN;
        sage_gemm_kernel<<<(total + 7) / 8, 256, 0, stream>>>(
            agg_drug, HID, HID, l1_dt_Wl, cnt_drug,
            h_drug,  HID, HID, l1_dt_Wr,
            l1_dt_bl, o_drug, OUT, tilesN, total, 0);
    }
}